// SelfAttentionBlock_24713241821416
// MI455X (gfx1250) — compile-verified
//
#include <hip/hip_runtime.h>

// SelfAttentionBlock (per-pixel channel attention), MI455X / gfx1250.
//
// Roofline: ~0.27 GFLOP vs ~604 MB streamed => 0.44 FLOP/byte, hard
// memory-bound (~26 us at 23.3 TB/s). WMMA is not applicable: every pixel
// has its own private 8x128 K / V matrix (no operand shared across a WMMA
// M/N dimension), so the CDNA5 feature that matters is the Tensor Data
// Mover: one tensor_load_to_lds pulls an (s=8, c=16, w=32) strided 3-D tile
// of k/v into LDS, double-buffered with s_wait_tensorcnt.
//
// One wave32 per workgroup: the TDM issuer is the consumer, so no barriers
// are needed at all — ordering is TENSORcnt-only. 2048 blocks, 32 KB LDS
// each (10 blocks/WGP).

#define N_   4
#define NS_  8
#define C_   128
#define H_   128
#define W_   128
#define TW   32              // pixels (w) per block = one wave32
#define CK   16              // channels per TDM chunk
#define NCH  (C_ / CK)       // 8 chunks
#define HW_  (H_ * W_)       // 16384: element stride between channels
#define CHW_ (C_ * H_ * W_)  // element stride between slots s
#define SCHW_ ((size_t)NS_ * C_ * H_ * W_)

typedef unsigned int u32;
typedef u32 u32x4 __attribute__((ext_vector_type(4)));
typedef u32 u32x8 __attribute__((ext_vector_type(8)));

// TDM load of a 3-D tile: dim0 = TW (w, contiguous), dim1 = CK (c, stride
// HW_), dim2 = NS_ (s, stride CHW_). Lands packed [s][c][w] f32 in LDS.
__device__ __forceinline__ void tdm_load_chunk(u32 lds_byte_addr, const float* gptr)
{
    unsigned long long ga = (unsigned long long)gptr;

    u32x4 g0;
    g0[0] = 1u;                                            // count=1 (valid)
    g0[1] = lds_byte_addr;                                 // LDS dest (bytes)
    g0[2] = (u32)ga;                                       // global_addr[31:0]
    g0[3] = ((u32)(ga >> 32) & 0x01ffffffu) | (2u << 30);  // addr[56:32] | type=2

    u32x8 g1;
    g1[0] = 2u << 16;                                      // data_size = 4B
    g1[1] = (u32)TW << 16;                                 // tensor_dim0[15:0]
    g1[2] = (u32)CK << 16;                                 // tensor_dim1[15:0]
    g1[3] = (u32)TW << 16;                                 // tile_dim0
    g1[4] = (u32)CK | ((u32)NS_ << 16);                    // tile_dim1 | tile_dim2
    g1[5] = (u32)HW_;                                      // tensor_dim0_stride[31:0]
    g1[6] = ((u32)CHW_ & 0xffffu) << 16;                   // tensor_dim1_stride[15:0] (=0)
    g1[7] = (u32)((unsigned long long)CHW_ >> 16);         // tensor_dim1_stride[47:16]

    u32x4 g2;
    g2[0] = (u32)NS_;                                      // tensor_dim2
    g2[1] = 0u; g2[2] = 0u; g2[3] = 0u;

    u32x4 g3;
    g3[0] = 0u; g3[1] = 0u; g3[2] = 0u; g3[3] = 0u;

    asm volatile("tensor_load_to_lds %0, %1, %2, %3"
                 :
                 : "s"(g0), "s"(g1), "s"(g2), "s"(g3)
                 : "memory");
}

// Waits as raw asm with a memory clobber: without workgroup barriers the
// LDS reads must not be hoisted above the tensor-count wait.
__device__ __forceinline__ void wait_tensorcnt_le1() {
    asm volatile("s_wait_tensorcnt 0x1" ::: "memory");
}
__device__ __forceinline__ void wait_tensorcnt_0() {
    asm volatile("s_wait_tensorcnt 0x0" ::: "memory");
}

__global__ __launch_bounds__(TW) void
SelfAttentionBlock_24713241821416_kernel(const float* __restrict__ q,
                                         const float* __restrict__ k,
                                         const float* __restrict__ v,
                                         float* __restrict__ out)
{
    __shared__ float lds[2][NS_ * CK * TW];                 // 2 x 16 KB

    const int tid = threadIdx.x;                            // w within tile
    const int bid = blockIdx.x;                             // 2048 blocks
    const int wt  = bid & (W_ / TW - 1);
    const int h   = (bid / (W_ / TW)) & (H_ - 1);
    const int n   = bid / ((W_ / TW) * H_);
    const int w0  = wt * TW;

    const float* qp    = q + ((size_t)n * CHW_ + (size_t)h * W_ + w0 + tid);
    const float* kbase = k + ((size_t)n * SCHW_ + (size_t)h * W_ + w0);
    const float* vbase = v + ((size_t)n * SCHW_ + (size_t)h * W_ + w0);
    float*       op    = out + ((size_t)n * CHW_ + (size_t)h * W_ + w0 + tid);

    const u32 lds_base = (u32)(size_t)(&lds[0][0]);
    const u32 BUFB     = (u32)(NS_ * CK * TW * sizeof(float));  // 16 KB

    // ---------------- Phase 1: scores[s] = sum_c q[c] * k[s][c] ----------
    float sc[NS_];
#pragma unroll
    for (int s = 0; s < NS_; ++s) sc[s] = 0.0f;

    tdm_load_chunk(lds_base, kbase);                        // chunk 0 -> buf 0

    for (int ci = 0; ci < NCH; ++ci) {
        // q loads issued BEFORE the wait so they overlap the DMA.
        float qc[CK];
#pragma unroll
        for (int c = 0; c < CK; ++c)
            qc[c] = __builtin_nontemporal_load(qp + (size_t)(ci * CK + c) * HW_);

        if (ci + 1 < NCH) {
            tdm_load_chunk(lds_base + ((u32)(ci + 1) & 1u) * BUFB,
                           kbase + (size_t)(ci + 1) * CK * HW_);
            wait_tensorcnt_le1();                           // chunk ci landed
        } else {
            wait_tensorcnt_0();
        }

        const float* kb = &lds[ci & 1][0];
#pragma unroll
        for (int c = 0; c < CK; ++c) {
#pragma unroll
            for (int s = 0; s < NS_; ++s)
                sc[s] = fmaf(qc[c], kb[(s * CK + c) * TW + tid], sc[s]);
        }
    }

    // Prefetch first v chunk; it overlaps the softmax math.
    tdm_load_chunk(lds_base, vbase);

    // ---------------- Softmax over s (scale 1/sqrt(128)) -----------------
    const float scale = 0.08838834764831845f;               // 1/sqrt(128)
    float m = sc[0];
#pragma unroll
    for (int s = 1; s < NS_; ++s) m = fmaxf(m, sc[s]);
    float al[NS_];
    float sum = 0.0f;
#pragma unroll
    for (int s = 0; s < NS_; ++s) {
        float e = __expf((sc[s] - m) * scale);
        al[s] = e;
        sum += e;
    }
    float inv = 1.0f / sum;
#pragma unroll
    for (int s = 0; s < NS_; ++s) al[s] *= inv;

    // ---------------- Phase 2: out[c] = sum_s alpha[s] * v[s][c] ---------
    for (int ci = 0; ci < NCH; ++ci) {
        if (ci + 1 < NCH) {
            tdm_load_chunk(lds_base + ((u32)(ci + 1) & 1u) * BUFB,
                           vbase + (size_t)(ci + 1) * CK * HW_);
            wait_tensorcnt_le1();
        } else {
            wait_tensorcnt_0();
        }

        const float* vb = &lds[ci & 1][0];
#pragma unroll
        for (int c = 0; c < CK; ++c) {
            float acc = 0.0f;
#pragma unroll
            for (int s = 0; s < NS_; ++s)
                acc = fmaf(al[s], vb[(s * CK + c) * TW + tid], acc);
            __builtin_nontemporal_store(acc, op + (size_t)(ci * CK + c) * HW_);
        }
    }
}

extern "C" void kernel_launch(void* const* d_in, const int* in_sizes, int n_in,
                              void* d_out, int out_size, void* d_ws, size_t ws_size,
                              hipStream_t stream)
{
    (void)in_sizes; (void)n_in; (void)out_size; (void)d_ws; (void)ws_size;
    const float* q = (const float*)d_in[0];
    const float* k = (const float*)d_in[1];
    const float* v = (const float*)d_in[2];
    float* out = (float*)d_out;

    dim3 grid(N_ * H_ * (W_ / TW));   // 2048 workgroups
    dim3 block(TW);                   // 32 threads = 1 wave32
    SelfAttentionBlock_24713241821416_kernel<<<grid, block, 0, stream>>>(q, k, v, out);
}